// WaveLayer_71597104824410
// MI455X (gfx1250) — compile-verified
//
#include <hip/hip_runtime.h>

// Problem constants (match reference setup_inputs exactly).
#define T_STEPS 400
#define NXG     128
#define NYG     128
#define DT2F    0.0025f          // DT*DT, DT=0.05 ; H=1 so no 1/h^2 scale

// LDS grid buffer: halo ring (rows/cols -1 and 128 stay zero forever) and an
// ODD row stride so that 32 consecutive-x lanes hit 32 distinct LDS banks.
#define SROW      131            // padded dwords per row (>=130, odd)
#define NROWS     130            // rows -1..128
#define BUF_ELEMS (NROWS * SROW) // 17030 floats per buffer
#define IDX(x, y) (((x) + 1) * SROW + ((y) + 1))

// One time step for this thread's 1x16 strip.
//   cur : LDS buffer holding y1 (read: neighbors only)
//   nxt : LDS buffer holding y2 on entry; overwritten with new y
//   y1  : register copy of own cells of y1
//   y2n : register copy of own cells of y2 on entry; holds new y on exit
__device__ __forceinline__ void step_body(
    const float* __restrict__ cur, float* __restrict__ nxt,
    float (&y1)[16], float (&y2n)[16],
    const float (&clin)[16], const float (&cnl)[16],
    int base, int sj, float srcAmp, float xt)
{
    const float lft = cur[base - 1];    // halo gives 0 at grid edge
    const float rgt = cur[base + 16];
#pragma unroll
    for (int q = 0; q < 4; ++q) {
        float up[4], dn[4];
#pragma unroll
        for (int k = 0; k < 4; ++k) up[k] = cur[base - SROW + q * 4 + k];
#pragma unroll
        for (int k = 0; k < 4; ++k) dn[k] = cur[base + SROW + q * 4 + k];
#pragma unroll
        for (int k = 0; k < 4; ++k) {
            const int j = q * 4 + k;
            const float l = (j == 0)  ? lft : y1[j - 1];
            const float r = (j == 15) ? rgt : y1[j + 1];
            const float lap = (up[k] + dn[k]) + (l + r) - 4.0f * y1[j];
            const float c   = clin[j] + (y1[j] * y1[j]) * cnl[j];
            float yn = 2.0f * y1[j] - y2n[j] + (DT2F * (c * c)) * lap;
            if (j == sj)                 // sj is scalar-uniform; srcAmp is 0
                yn += srcAmp * xt;       // except on the single owner thread
            y2n[j] = yn;
            nxt[base + j] = yn;
        }
    }
    __syncthreads();                     // publish nxt; retire reads of cur
}

extern "C" __global__ void __launch_bounds__(1024, 1)
wave_main(const float* __restrict__ inputs,   // [B, T]
          const float* __restrict__ c_lin,    // [128,128]
          const float* __restrict__ c_nl,     // [128,128]
          const int*   __restrict__ psrc_x,
          const int*   __restrict__ psrc_y,
          const int*   __restrict__ probes,   // [3,2]
          float*       __restrict__ partial)  // [B,3] probe energy partials
{
    extern __shared__ float smem[];
    float* xin  = smem;                 // 400 floats, 16B aligned (offset 0)
    float* bufA = smem + T_STEPS;       // y buffers, ping-pong
    float* bufB = bufA + BUF_ELEMS;

    const int tid = threadIdx.x;        // 1024 threads = 32 wave32 waves
    const int b   = blockIdx.x;         // one batch per workgroup / WGP
    const int x   = tid & 127;          // lane-consecutive rows
    const int cb  = tid >> 7;           // column block 0..7
    const int y0  = cb << 4;
    const int base = IDX(x, y0);

    // ---- CDNA5 async global->LDS copy of this batch's input waveform ----
    // 400 floats = 100 lanes x b128. Tracked on ASYNCcnt, overlaps the
    // LDS zero-fill below.
    if (tid < T_STEPS / 4) {
        const float* gsrc = inputs + (size_t)b * T_STEPS + tid * 4;
        const unsigned ldst = (unsigned)(size_t)(xin + tid * 4); // LDS offset
        asm volatile("global_load_async_to_lds_b128 %0, %1, off"
                     :: "v"(ldst), "v"(gsrc)
                     : "memory");
    }

    // Zero both grid buffers (incl. halo ring, which is never written again).
    for (int i = tid; i < 2 * BUF_ELEMS; i += 1024) bufA[i] = 0.0f;

    // Per-thread coefficient registers.
    float clin[16], cnl[16], ya[16], yb[16];
#pragma unroll
    for (int j = 0; j < 16; ++j) {
        clin[j] = c_lin[x * NYG + y0 + j];
        cnl[j]  = c_nl [x * NYG + y0 + j];
        ya[j] = 0.0f;                   // y1 = y0 = 0
        yb[j] = 0.0f;                   // y2 = 0
    }

    // Source ownership: exactly one thread has srcAmp != 0.
    const int sx = psrc_x[0];
    const int sy = psrc_y[0];
    const int sj = sy & 15;             // scalar-uniform
    const float srcAmp = (x == sx && (sy >> 4) == cb) ? DT2F : 0.0f;

    // Probe readers: threads 0..2.
    int paddr = 0;
    const bool pvalid = (tid < 3);
    if (pvalid) paddr = IDX(probes[tid * 2], probes[tid * 2 + 1]);
    float acc = 0.0f;

    asm volatile("s_wait_asynccnt 0" ::: "memory");  // xin resident in LDS
    __syncthreads();                                  // zeros + xin visible

    // ---- 400 fully on-chip time steps, unrolled by 2 (ping-pong) ----
    for (int t = 0; t < T_STEPS; t += 2) {
        step_body(bufA, bufB, ya, yb, clin, cnl, base, sj, srcAmp, xin[t]);
        if (pvalid) { const float v = bufB[paddr]; acc += v * v; }
        step_body(bufB, bufA, yb, ya, clin, cnl, base, sj, srcAmp, xin[t + 1]);
        if (pvalid) { const float v = bufA[paddr]; acc += v * v; }
    }

    if (pvalid) partial[b * 3 + tid] = acc;   // deterministic, no atomics
}

extern "C" __global__ void wave_finalize(const float* __restrict__ partial,
                                         float* __restrict__ out, int nb)
{
    if (threadIdx.x == 0) {
        float s0 = 0.0f, s1 = 0.0f, s2 = 0.0f;
        for (int b = 0; b < nb; ++b) {
            s0 += partial[b * 3 + 0];
            s1 += partial[b * 3 + 1];
            s2 += partial[b * 3 + 2];
        }
        const float inv = 1.0f / (s0 + s1 + s2);
        out[0] = s0 * inv; out[1] = s1 * inv; out[2] = s2 * inv;
    }
}

extern "C" void kernel_launch(void* const* d_in, const int* in_sizes, int n_in,
                              void* d_out, int out_size, void* d_ws, size_t ws_size,
                              hipStream_t stream)
{
    const float* inputs = (const float*)d_in[0];  // [B,400] f32
    const float* c_lin  = (const float*)d_in[1];  // [128,128] f32
    const float* c_nl   = (const float*)d_in[2];  // [128,128] f32
    const int*   src_x  = (const int*)d_in[3];
    const int*   src_y  = (const int*)d_in[4];
    const int*   probes = (const int*)d_in[5];    // [3,2] i32

    const int B = in_sizes[0] / T_STEPS;          // 16
    float* partial = (float*)d_ws;                // B*3 floats of scratch

    const size_t shmem = (size_t)(T_STEPS + 2 * BUF_ELEMS) * sizeof(float); // ~136 KB
    wave_main<<<B, 1024, shmem, stream>>>(inputs, c_lin, c_nl,
                                          src_x, src_y, probes, partial);
    wave_finalize<<<1, 32, 0, stream>>>(partial, (float*)d_out, B);
}